// BiologicalNormalization_60447369724342
// MI455X (gfx1250) — compile-verified
//
#include <hip/hip_runtime.h>

#define DD 512
#define SS 128
#define LN_EPS 1e-5f

typedef float v4f __attribute__((ext_vector_type(4)));
typedef int   v4i __attribute__((ext_vector_type(4)));

#if defined(__HIP_DEVICE_COMPILE__)
#if __has_builtin(__builtin_amdgcn_global_load_async_to_lds_b128)
#define HAVE_ASYNC_LDS 1
#endif
#endif

// Copy 16 bytes global -> LDS. Prefer the gfx1250 async-to-LDS path
// (tracked by ASYNCcnt), fall back to a plain register round-trip.
// The builtin takes (int4* global_src, int4* lds_dst, imm offset, imm cpol);
// generic pointers are accepted and codegen inserts the addrspacecasts.
__device__ __forceinline__ void async_cp16(const float* src, float* lds_dst) {
#if defined(HAVE_ASYNC_LDS)
    __builtin_amdgcn_global_load_async_to_lds_b128(
        (v4i*)src, (v4i*)lds_dst, /*offset=*/0, /*cpol=*/0);
#else
    *(v4f*)lds_dst = *(const v4f*)src;
#endif
}

__device__ __forceinline__ void async_wait_all() {
#if defined(HAVE_ASYNC_LDS)
#if __has_builtin(__builtin_amdgcn_s_wait_asynccnt)
    __builtin_amdgcn_s_wait_asynccnt(0);
#else
    asm volatile("s_wait_asynccnt 0" ::: "memory");
#endif
#endif
}

__global__ __launch_bounds__(256) void bionorm_kernel(
    const float* __restrict__ x,
    const int* __restrict__ pid,
    const int* __restrict__ cid,
    const int* __restrict__ tid,
    const float* __restrict__ pg, const float* __restrict__ pb,
    const float* __restrict__ cg, const float* __restrict__ cb,
    const float* __restrict__ tg, const float* __restrict__ tb,
    float* __restrict__ out)
{
    __shared__ float lds[6 * DD];   // [g0,b0,g1,b1,g2,b2] x 512 floats = 12 KiB

    const int b    = blockIdx.x;
    const int t    = threadIdx.x;
    const int lane = t & 31;
    const int wave = t >> 5;        // 0..7

    // Per-sample id gathers (uniform per block -> scalar loads)
    const int p  = pid[b];
    const int c  = cid[b];
    const int ct = tid[b];

    // Stage the 6 affine vectors into LDS with async-to-LDS copies.
    // 6*512 floats = 768 float4s; 256 threads x 3 copies.
    {
        const int half = t >> 7;           // 0 = gamma half, 1 = beta half
        const int e4   = (t & 127) << 2;   // float offset within vector (16B aligned)
        const float* g0 = pg + (size_t)p  * DD;
        const float* b0 = pb + (size_t)p  * DD;
        const float* g1 = cg + (size_t)c  * DD;
        const float* b1 = cb + (size_t)c  * DD;
        const float* g2 = tg + (size_t)ct * DD;
        const float* b2 = tb + (size_t)ct * DD;
        async_cp16((half ? b0 : g0) + e4, &lds[(0 + half) * DD + e4]);
        async_cp16((half ? b1 : g1) + e4, &lds[(2 + half) * DD + e4]);
        async_cp16((half ? b2 : g2) + e4, &lds[(4 + half) * DD + e4]);
    }
    async_wait_all();
    __syncthreads();

    const size_t sampleBase = (size_t)b * SS * DD;

    // Each wave owns rows s = wave, wave+8, ... (16 rows). Row stays in
    // 16 VGPRs/lane across all three chained LayerNorms: x is read once.
    for (int s = wave; s < SS; s += 8) {
        const float* row  = x   + sampleBase + (size_t)s * DD;
        float*       orow = out + sampleBase + (size_t)s * DD;

        if (s + 8 < SS)
            __builtin_prefetch(row + 8 * DD, 0, 1);   // global_prefetch_b8

        v4f v[4];
        #pragma unroll
        for (int k = 0; k < 4; ++k)
            v[k] = *(const v4f*)(row + k * 128 + lane * 4);

        #pragma unroll
        for (int ln = 0; ln < 3; ++ln) {
            // mean / mean-of-squares over the 512-element row
            float sum = 0.0f, ssq = 0.0f;
            #pragma unroll
            for (int k = 0; k < 4; ++k) {
                #pragma unroll
                for (int j = 0; j < 4; ++j) {
                    const float f = v[k][j];
                    sum += f;
                    ssq  = fmaf(f, f, ssq);
                }
            }
            // wave32 butterfly reduction (5 steps)
            #pragma unroll
            for (int m = 16; m >= 1; m >>= 1) {
                sum += __shfl_xor(sum, m, 32);
                ssq += __shfl_xor(ssq, m, 32);
            }
            const float mean = sum * (1.0f / DD);
            const float var  = fmaf(-mean, mean, ssq * (1.0f / DD));
            const float rs   = rsqrtf(var + LN_EPS);

            const float* gl = &lds[(2 * ln + 0) * DD];
            const float* bl = &lds[(2 * ln + 1) * DD];
            #pragma unroll
            for (int k = 0; k < 4; ++k) {
                const v4f gv = *(const v4f*)(gl + k * 128 + lane * 4);
                const v4f bv = *(const v4f*)(bl + k * 128 + lane * 4);
                #pragma unroll
                for (int j = 0; j < 4; ++j)
                    v[k][j] = fmaf((v[k][j] - mean) * rs, gv[j], bv[j]);
            }
        }

        #pragma unroll
        for (int k = 0; k < 4; ++k)
            *(v4f*)(orow + k * 128 + lane * 4) = v[k];
    }
}

extern "C" void kernel_launch(void* const* d_in, const int* in_sizes, int n_in,
                              void* d_out, int out_size, void* d_ws, size_t ws_size,
                              hipStream_t stream) {
    (void)n_in; (void)out_size; (void)d_ws; (void)ws_size;
    const float* x  = (const float*)d_in[0];
    const int* pid  = (const int*)d_in[1];
    const int* cid  = (const int*)d_in[2];
    const int* tid  = (const int*)d_in[3];
    const float* pg = (const float*)d_in[4];
    const float* pb = (const float*)d_in[5];
    const float* cg = (const float*)d_in[6];
    const float* cb = (const float*)d_in[7];
    const float* tg = (const float*)d_in[8];
    const float* tb = (const float*)d_in[9];
    // d_in[10] = W, d_in[11] = b are intentionally unused:
    // x*sigmoid(xW+b) + x*(1-sigmoid(xW+b)) == x, so the gate GEMM is dead math.
    float* out = (float*)d_out;

    const int B = in_sizes[1];     // pathway_ids has one entry per sample
    hipLaunchKernelGGL(bionorm_kernel, dim3(B), dim3(256), 0, stream,
                       x, pid, cid, tid, pg, pb, cg, cb, tg, tb, out);
}